// LennardJonesPotential_34385508172081
// MI455X (gfx1250) — compile-verified
//
#include <hip/hip_runtime.h>
#include <math.h>

// CDNA5 / gfx1250: LJ potential via WMMA Gram matrix, symmetric-tile version.
// d2_ij = n_i + n_j - 2*G_ij,  G = X X^T computed with v_wmma_f32_16x16x4_f32.
// Only lower-triangle tiles processed; off-diagonal tiles weighted 2x.

typedef float v2f __attribute__((ext_vector_type(2)));
typedef float v8f __attribute__((ext_vector_type(8)));

#define NP      512
#define ND      3
#define TILES16 32                          // 512 / 16
#define NTRI    (TILES16 * (TILES16 + 1) / 2)   // 528 lower-triangle tiles
#define BLOCK   256                         // 8 wave32s
#define NWAVES  (BLOCK / 32)                // 528 = 8 * 66 -> perfectly balanced
#define DIST_EPS 1e-6f

__global__ __launch_bounds__(BLOCK)
void LennardJonesPotential_kernel(const float* __restrict__ xg,
                                  float* __restrict__ out) {
  __shared__ float xs[NP];
  __shared__ float ys[NP];
  __shared__ float zs[NP];
  __shared__ float nrm[NP];
  __shared__ float red[BLOCK];

  const int b   = blockIdx.x;
  const int tid = threadIdx.x;
  const float* __restrict__ xb = xg + (size_t)b * (NP * ND);

  // ---- Stage coordinates into LDS (SoA), coalesced global reads ----
  for (int i = tid; i < NP * ND; i += BLOCK) {
    float v = xb[i];
    int p = i / 3;
    int d = i - 3 * p;
    if (d == 0)      xs[p] = v;
    else if (d == 1) ys[p] = v;
    else             zs[p] = v;
  }
  __syncthreads();

  // ---- Norms + per-thread partial sums for the harmonic term ----
  float psx = 0.f, psy = 0.f, psz = 0.f, pss = 0.f;
  for (int p = tid; p < NP; p += BLOCK) {
    float x = xs[p], y = ys[p], z = zs[p];
    float n = x * x + y * y + z * z;
    nrm[p] = n;
    psx += x; psy += y; psz += z; pss += n;
  }
  __syncthreads();

  // ---- Gram tiles via WMMA f32 16x16x4 (K: x,y,z,0) + LJ post-process ----
  const int  lane = tid & 31;
  const int  wave = tid >> 5;
  const int  m    = lane & 15;       // M for A-frag, N for B-frag, N for C
  const bool hi   = lane >= 16;      // selects K=2,3 half for A/B frags

  float acc = 0.0f;

  for (int t = wave; t < NTRI; t += NWAVES) {  // 66 iters per wave, uniform
    // Invert triangular index: i*(i+1)/2 <= t < (i+1)*(i+2)/2, j = remainder
    int i = (int)((sqrtf(8.0f * (float)t + 1.0f) - 1.0f) * 0.5f);
    while ((i + 1) * (i + 2) / 2 <= t) ++i;    // guard fp rounding
    while (i * (i + 1) / 2 > t)       --i;
    const int j  = t - i * (i + 1) / 2;        // j <= i
    const int tr = i << 4;                     // row tile base (particle idx)
    const int tc = j << 4;                     // col tile base

    // A frag: row = tr+m; VGPR0 = K0(x) / K2(z), VGPR1 = K1(y) / K3(pad 0)
    v2f a, bmat;
    a.x    = hi ? zs[tr + m] : xs[tr + m];
    a.y    = hi ? 0.0f       : ys[tr + m];
    // B frag (X^T, col = tc+m): same structure with N across lanes 0-15
    bmat.x = hi ? zs[tc + m] : xs[tc + m];
    bmat.y = hi ? 0.0f       : ys[tc + m];

    v8f c = {};
    c = __builtin_amdgcn_wmma_f32_16x16x4_f32(
            /*neg_a=*/false, a, /*neg_b=*/false, bmat,
            /*c_mod=*/(short)0, c, /*reuse_a=*/false, /*reuse_b=*/false);

    // C/D layout: lane&15 = N, VGPR v -> M = v + 8*(lane>=16)
    const int   col   = tc + m;
    const float ncol  = nrm[col];
    const int   rbase = tr + (hi ? 8 : 0);

    float tacc = 0.0f;
#pragma unroll
    for (int v = 0; v < 8; ++v) {
      const int row = rbase + v;
      float d2   = nrm[row] + ncol - 2.0f * c[v] + DIST_EPS;
      float inv  = __builtin_amdgcn_rcpf(d2);      // v_rcp_f32
      float inv3 = inv * inv * inv;                // (1/r)^6 = (d2+eps)^-3
      float lj   = inv3 * inv3 - 2.0f * inv3;
      tacc += (row != col) ? lj : 0.0f;            // mask diagonal
    }
    // off-diagonal tiles account for both (i,j) and (j,i)
    acc = fmaf((i == j) ? 1.0f : 2.0f, tacc, acc);
  }

  // ---- Deterministic block reductions (no float atomics) ----
  auto block_reduce = [&](float v) -> float {
    red[tid] = v;
    __syncthreads();
    float r = 0.0f;
    if (tid == 0) {
      float s = 0.0f;
#pragma unroll 8
      for (int i2 = 0; i2 < BLOCK; ++i2) s += red[i2];
      r = s;
    }
    __syncthreads();
    return r;          // meaningful on tid 0 only
  };

  float lj_tot = block_reduce(acc);
  float sx     = block_reduce(psx);
  float sy     = block_reduce(psy);
  float sz     = block_reduce(psz);
  float ss     = block_reduce(pss);

  if (tid == 0) {
    // 0.5 * sum |x - mean|^2 = 0.5 * (sum|x|^2 - |sum x|^2 / N)
    float osc = 0.5f * (ss - (sx * sx + sy * sy + sz * sz) * (1.0f / NP));
    out[b] = lj_tot + osc;   // ENERGY_FACTOR = OSC_SCALE = 1
  }
}

extern "C" void kernel_launch(void* const* d_in, const int* in_sizes, int n_in,
                              void* d_out, int out_size, void* d_ws, size_t ws_size,
                              hipStream_t stream) {
  const float* x = (const float*)d_in[0];
  float* out     = (float*)d_out;
  const int B    = in_sizes[0] / (NP * ND);   // 128
  LennardJonesPotential_kernel<<<B, BLOCK, 0, stream>>>(x, out);
}